// MLP_subgraph_34711925686946
// MI455X (gfx1250) — compile-verified
//
#include <hip/hip_runtime.h>

typedef __attribute__((ext_vector_type(2))) float v2f;
typedef __attribute__((ext_vector_type(8))) float v8f;

#define F_IN 128
#define FS   64
#define H    8

// ---------------------------------------------------------------------------
// Kernel 1: x64[N,64] = x[N,128] @ Wlin[64,128]^T + blin  via V_WMMA_F32_16X16X4_F32
// One 16x16 output tile per wave; K-loop 128/4 = 32 WMMAs.
// A layout (16x4 f32): lanes 0-15 -> M=lane, K={k0,k0+1}; lanes 16-31 -> K={k0+2,k0+3}
// B layout (4x16 f32): B[k][n] = Wlin[n][k], lane%16 = n, same K split.
// D layout: VGPR v -> M = v + (lane<16?0:8), N = lane%16.
// ---------------------------------------------------------------------------
__global__ __launch_bounds__(256) void gemm_x64_kernel(
    const float* __restrict__ x, const float* __restrict__ Wlin,
    const float* __restrict__ blin, float* __restrict__ x64) {
  const int lane = threadIdx.x & 31;
  const int wave = threadIdx.x >> 5;
  const int tile = blockIdx.x * 8 + wave;   // Mtiles*Ntiles total
  const int tm   = tile >> 2;               // 4 N-tiles (FS/16)
  const int tn   = tile & 3;
  const int hs   = lane >> 4;               // half select
  const int lm   = lane & 15;
  const int koff = hs * 2;

  const float* xrow = x    + (tm * 16 + lm) * F_IN;  // A row (node)
  const float* wrow = Wlin + (tn * 16 + lm) * F_IN;  // B col n = Wlin row

  v8f acc = {};
  #pragma unroll 4
  for (int k0 = 0; k0 < F_IN; k0 += 4) {
    v2f a = *(const v2f*)(xrow + k0 + koff);
    v2f b = *(const v2f*)(wrow + k0 + koff);
    acc = __builtin_amdgcn_wmma_f32_16x16x4_f32(
        /*neg_a=*/false, a, /*neg_b=*/false, b,
        /*c_mod=*/(short)0, acc, /*reuse_a=*/false, /*reuse_b=*/false);
  }

  const int ncol  = tn * 16 + lm;
  const float bv  = blin[ncol];
  const int mbase = tm * 16 + hs * 8;
  #pragma unroll
  for (int v = 0; v < 8; ++v)
    x64[(mbase + v) * FS + ncol] = acc[v] + bv;
}

// ---------------------------------------------------------------------------
// Kernel 2: hcat[N,16] = x64[N,64] @ W1cat[16,64]^T, where
//   W1cat[h][k] = W1[h][k]      for h<8   (h1 part)
//   W1cat[h][k] = W1[h-8][64+k] for h>=8  (h2 part)
// N dimension is exactly one 16-wide tile; K-loop 64/4 = 16 WMMAs.
// ---------------------------------------------------------------------------
__global__ __launch_bounds__(256) void gemm_hcat_kernel(
    const float* __restrict__ x64, const float* __restrict__ W1,
    float* __restrict__ hcat) {
  const int lane = threadIdx.x & 31;
  const int wave = threadIdx.x >> 5;
  const int tm   = blockIdx.x * 8 + wave;
  const int hs   = lane >> 4;
  const int lm   = lane & 15;
  const int koff = hs * 2;

  const float* xrow = x64 + (tm * 16 + lm) * FS;
  const float* wrow = W1 + (lm & 7) * (2 * FS) + (lm >> 3) * FS;

  v8f acc = {};
  #pragma unroll 4
  for (int k0 = 0; k0 < FS; k0 += 4) {
    v2f a = *(const v2f*)(xrow + k0 + koff);
    v2f b = *(const v2f*)(wrow + k0 + koff);
    acc = __builtin_amdgcn_wmma_f32_16x16x4_f32(
        false, a, false, b, (short)0, acc, false, false);
  }

  const int mbase = tm * 16 + hs * 8;
  #pragma unroll
  for (int v = 0; v < 8; ++v)
    hcat[(mbase + v) * 16 + lm] = acc[v];
}

// ---------------------------------------------------------------------------
// Kernel 3: per-edge gate.
//   s_ij = sig(b2 + sum_h W2[h]*sig(h1[i,h]+h2[j,h]+b1[h]))
//   w = 0.5*(s_ij + s_ji);  gate = sig(2*(log(eps)-log1p(-eps)+w))
// ---------------------------------------------------------------------------
__device__ __forceinline__ float sigf(float z) {
  return 1.0f / (1.0f + __expf(-z));
}

__global__ __launch_bounds__(256) void edge_gate_kernel(
    const float* __restrict__ hcat, const int* __restrict__ edge_index,
    const float* __restrict__ u, const float* __restrict__ b1v,
    const float* __restrict__ W2, const float* __restrict__ b2,
    float* __restrict__ gate, int E) {
  __shared__ float sb1[H], sW2[H], sb2;
  if (threadIdx.x < H) { sb1[threadIdx.x] = b1v[threadIdx.x]; sW2[threadIdx.x] = W2[threadIdx.x]; }
  if (threadIdx.x == 0) sb2 = b2[0];
  __syncthreads();

  const int e = blockIdx.x * blockDim.x + threadIdx.x;
  if (e >= E) return;

  const int i = edge_index[e];
  const int j = edge_index[E + e];
  const float* hi = hcat + i * 16;  // [0:8]=h1[i], [8:16]=h2[i]
  const float* hj = hcat + j * 16;

  float hib[16], hjb[16];
  #pragma unroll
  for (int t = 0; t < 16; ++t) { hib[t] = hi[t]; hjb[t] = hj[t]; }

  float aij = sb2, aji = sb2;
  #pragma unroll
  for (int h = 0; h < H; ++h) {
    aij += sW2[h] * sigf(hib[h] + hjb[8 + h] + sb1[h]);
    aji += sW2[h] * sigf(hjb[h] + hib[8 + h] + sb1[h]);
  }
  const float w = 0.5f * (sigf(aij) + sigf(aji));

  const float bias = 1.0e-4f;
  const float uu   = u[e];
  const float eps  = (2.0f * bias - 1.0f) * uu + (1.0f - bias);
  const float lgt  = __logf(eps) - log1pf(-eps);
  gate[e] = sigf(2.0f * (lgt + w));   // temperature 0.5
}

// ---------------------------------------------------------------------------
// Kernel 4: unbiased variance of gate[0..E) -> out[0]. Single 1024-thread block.
// ---------------------------------------------------------------------------
__global__ __launch_bounds__(1024) void var_kernel(
    const float* __restrict__ gate, int E, float* __restrict__ out) {
  __shared__ float ssum[1024];
  __shared__ float ssq[1024];
  float s = 0.0f, sq = 0.0f;
  for (int e = threadIdx.x; e < E; e += 1024) {
    float g = gate[e];
    s += g; sq += g * g;
  }
  ssum[threadIdx.x] = s; ssq[threadIdx.x] = sq;
  __syncthreads();
  for (int stride = 512; stride > 0; stride >>= 1) {
    if (threadIdx.x < stride) {
      ssum[threadIdx.x] += ssum[threadIdx.x + stride];
      ssq[threadIdx.x]  += ssq[threadIdx.x + stride];
    }
    __syncthreads();
  }
  if (threadIdx.x == 0) {
    const float n = (float)E;
    out[0] = (ssq[0] - ssum[0] * ssum[0] / n) / (n - 1.0f);
  }
}

// ---------------------------------------------------------------------------
extern "C" void kernel_launch(void* const* d_in, const int* in_sizes, int n_in,
                              void* d_out, int out_size, void* d_ws, size_t ws_size,
                              hipStream_t stream) {
  const float* x    = (const float*)d_in[0];
  const float* Wlin = (const float*)d_in[1];
  const float* blin = (const float*)d_in[2];
  const float* W1   = (const float*)d_in[3];
  const float* b1   = (const float*)d_in[4];
  const float* W2   = (const float*)d_in[5];
  const float* b2   = (const float*)d_in[6];
  const int* edge_index = (const int*)d_in[7];
  const float* u    = (const float*)d_in[8];

  const int N = in_sizes[0] / F_IN;   // 4096
  const int E = in_sizes[8];          // 131072

  float* gate = (float*)d_out;        // d_out = [gate(E), pos_penalty(1)]
  float* x64  = (float*)d_ws;         // N*FS floats
  float* hcat = x64 + (size_t)N * FS; // N*16 floats

  // Kernel 1: (N/16)*(FS/16) tiles, 8 waves (tiles) per block.
  {
    const int tiles = (N / 16) * (FS / 16);    // 1024
    gemm_x64_kernel<<<tiles / 8, 256, 0, stream>>>(x, Wlin, blin, x64);
  }
  // Kernel 2: N/16 tiles along M, single N-tile.
  {
    const int tiles = N / 16;                  // 256
    gemm_hcat_kernel<<<tiles / 8, 256, 0, stream>>>(x64, W1, hcat);
  }
  // Kernel 3: one thread per edge.
  edge_gate_kernel<<<(E + 255) / 256, 256, 0, stream>>>(
      hcat, edge_index, u, b1, W2, b2, gate, E);
  // Kernel 4: variance -> d_out[E].
  var_kernel<<<1, 1024, 0, stream>>>(gate, E, gate + E);
}